// Window_Attention_76252849373618
// MI455X (gfx1250) — compile-verified
//
#include <hip/hip_runtime.h>
#include <hip/hip_bf16.h>

// ---------------- problem constants ----------------
#define WS7     7
#define TOK     49          // tokens per window
#define HEADS   3
#define HD      32
#define HIDDEN  96
#define INNER   96          // heads*hd
#define QKVN    288         // 3*inner
#define NWH     8           // windows per side
#define NW      64          // windows per image
#define BATCH   32
#define IMGH    56
#define IMGW    56
#define DISP    3
#define SCALE_Q 0.10206207261596577f          // 96^-0.5
#define PERMAT  9633792                        // BATCH*HEADS*NW*TOK*HD floats

typedef __attribute__((ext_vector_type(2))) float v2f;
typedef __attribute__((ext_vector_type(8))) float v8f;

// V_WMMA_F32_16X16X4_F32 : D = A(16x4) * B(4x16) + C(16x16), fp32 throughout.
__device__ __forceinline__ v8f wmma4(v2f a, v2f b, v8f c) {
    return __builtin_amdgcn_wmma_f32_16x16x4_f32(
        /*neg_a=*/false, a, /*neg_b=*/false, b,
        /*c_mod=*/(short)0, c, /*reuse_a=*/false, /*reuse_b=*/false);
}

// =====================================================================
// Kernel 1: shifted-gather QKV projection -> windowed q/k/v layout
//   grid (1568, 6) x block 128 : 4 waves/block, one 16x48 tile per wave
// =====================================================================
__global__ __launch_bounds__(128) void qkv_kernel(
    const float* __restrict__ x, const float* __restrict__ wqkv,
    const float* __restrict__ bqkv,
    float* __restrict__ qws, float* __restrict__ kws, float* __restrict__ vws) {

    const int lane  = threadIdx.x & 31;
    const int nlane = lane & 15;
    const int khalf = lane >> 4;
    const int mt    = blockIdx.x * 4 + (threadIdx.x >> 5);   // M-tile (16 tokens)
    const int nbase = blockIdx.y * 48;

    // gather: this lane supplies A rows for token (mt*16 + nlane)
    int tglob = mt * 16 + nlane;
    int p  = tglob % TOK;
    int wi = (tglob / TOK) % NW;
    int bb = tglob / (TOK * NW);
    int i = p / WS7, j = p % WS7;
    int wr = wi / NWH, wc = wi % NWH;
    int hs = wr * WS7 + i + DISP;  if (hs >= IMGH) hs -= IMGH;
    int ww = wc * WS7 + j + DISP;  if (ww >= IMGW) ww -= IMGW;
    const float* arow = x + ((size_t)((bb * IMGH + hs) * IMGW + ww)) * HIDDEN;

    v8f c0 = {}, c1 = {}, c2 = {};
#pragma unroll
    for (int k0 = 0; k0 < HIDDEN; k0 += 4) {
        const int kk = k0 + 2 * khalf;
        v2f a = *(const v2f*)(arow + kk);                       // A: [m][kk..kk+1]
        const float* wrow = wqkv + (size_t)kk * QKVN + nbase + nlane;
        v2f b0; b0.x = wrow[0];  b0.y = wrow[QKVN];
        v2f b1; b1.x = wrow[16]; b1.y = wrow[QKVN + 16];
        v2f b2; b2.x = wrow[32]; b2.y = wrow[QKVN + 32];
        c0 = wmma4(a, b0, c0);
        c1 = wmma4(a, b1, c1);
        c2 = wmma4(a, b2, c2);
    }

    float* const bases[3] = { qws, kws, vws };
#pragma unroll
    for (int nt = 0; nt < 3; ++nt) {
        const int col  = nbase + nt * 16 + nlane;
        const int mat  = col / INNER;            // 0=q 1=k 2=v (uniform per tile)
        const int cc   = col % INNER;
        const int head = cc >> 5, d = cc & 31;
        const float bias = bqkv[col];
        v8f acc = (nt == 0) ? c0 : ((nt == 1) ? c1 : c2);
#pragma unroll
        for (int r = 0; r < 8; ++r) {
            int tok = mt * 16 + 8 * khalf + r;   // C layout: M = 8*half + r
            int pp  = tok % TOK;
            int wn  = (tok / TOK) % NW;
            int b2  = tok / (TOK * NW);
            float v = acc[r] + bias;
            if (mat == 0) v *= SCALE_Q;          // fold softmax scale into q
            bases[mat][((((size_t)b2 * HEADS + head) * NW + wn) * TOK + pp) * HD + d] = v;
        }
    }
}

// =====================================================================
// Kernel 2: per-(b,head,window) attention. grid 6144 x block 128.
//   wave w handles rows [16w,16w+16) of the 49x49 problem.
//   Output aliases the q buffer (each wave only rereads/writes its rows).
// =====================================================================
__global__ __launch_bounds__(128) void attn_kernel(
    const float* __restrict__ qws, const float* __restrict__ kws,
    const float* __restrict__ vws, const float* __restrict__ pos,
    float* __restrict__ ows) {

    __shared__ float sK[64][33];        // padded to kill bank conflicts
    __shared__ float sV[64][33];
    __shared__ float sAtt[4][16][66];   // even stride -> 8B-aligned v2f loads
    __shared__ float sPos[169];

    const int pid  = blockIdx.x;                  // ((b*3+head)*64+win)
    const int win  = pid & 63;
    const size_t base = (size_t)pid * (TOK * HD);
    const float* Q = qws + base;
    const float* K = kws + base;
    const float* V = vws + base;
    float*       O = ows + base;

    for (int idx = threadIdx.x; idx < 64 * 32; idx += 128) {
        int rr = idx >> 5, cc = idx & 31;
        float kv = 0.f, vv = 0.f;
        if (rr < TOK) { kv = K[rr * HD + cc]; vv = V[rr * HD + cc]; }
        sK[rr][cc] = kv;
        sV[rr][cc] = vv;
    }
    if (threadIdx.x < 169) sPos[threadIdx.x] = pos[threadIdx.x];
    __syncthreads();

    const int wave  = threadIdx.x >> 5;
    const int lane  = threadIdx.x & 31;
    const int nlane = lane & 15;
    const int khalf = lane >> 4;
    const int arow  = wave * 16 + nlane;          // A row this lane feeds
    const bool rowok = arow < TOK;

    // ---- logits = q @ k^T (q pre-scaled) ----
    v8f acc[4] = {};
#pragma unroll
    for (int k0 = 0; k0 < HD; k0 += 4) {
        const int kk = k0 + 2 * khalf;
        v2f a;
        if (rowok) a = *(const v2f*)(Q + (size_t)arow * HD + kk);
        else       { a.x = 0.f; a.y = 0.f; }
#pragma unroll
        for (int nt = 0; nt < 4; ++nt) {
            const int n = nt * 16 + nlane;
            v2f b; b.x = sK[n][kk]; b.y = sK[n][kk + 1];
            acc[nt] = wmma4(a, b, acc[nt]);
        }
    }

    // ---- bias + shifted mask + row softmax (C-layout rows live on a half-wave) ----
    const bool shifted = (win >= (NW - NWH));     // last window-row block
    const float NEGINF = -__builtin_inff();
#pragma unroll
    for (int r = 0; r < 8; ++r) {
        const int irow = wave * 16 + 8 * khalf + r;
        float vals[4];
        float rmax = -3.0e38f;
#pragma unroll
        for (int nt = 0; nt < 4; ++nt) {
            const int jcol = nt * 16 + nlane;
            float v = acc[nt][r];
            if (jcol >= TOK)      v = NEGINF;     // padded columns
            else if (irow >= TOK) v = 0.f;        // padded rows: keep finite (unused)
            else {
                int ri = (jcol / WS7) - (irow / WS7) + (WS7 - 1);
                int rj = (jcol % WS7) - (irow % WS7) + (WS7 - 1);
                v += sPos[ri * 13 + rj];
                if (shifted && ((irow >= DISP * WS7 + WS7) != (jcol >= DISP * WS7 + WS7))) {
                    // 28 = WS*WS - DISP*WS
                }
                if (shifted && ((irow >= 28) != (jcol >= 28))) v = NEGINF;
            }
            vals[nt] = v;
            rmax = fmaxf(rmax, v);
        }
#pragma unroll
        for (int s = 1; s < 16; s <<= 1)
            rmax = fmaxf(rmax, __shfl_xor(rmax, s, 16));
        float rsum = 0.f;
#pragma unroll
        for (int nt = 0; nt < 4; ++nt) {
            float e = __expf(vals[nt] - rmax);
            vals[nt] = e;
            rsum += e;
        }
#pragma unroll
        for (int s = 1; s < 16; s <<= 1)
            rsum += __shfl_xor(rsum, s, 16);
        const float rinv = 1.0f / rsum;
        const int mr = 8 * khalf + r;
#pragma unroll
        for (int nt = 0; nt < 4; ++nt)
            sAtt[wave][mr][nt * 16 + nlane] = vals[nt] * rinv;
    }
    __syncthreads();   // uniform; orders sAtt stores before A-layout reloads

    // ---- out = att @ v  (K padded to 52; padded att cols/V rows are zero) ----
    v8f oacc[2] = {};
#pragma unroll
    for (int k0 = 0; k0 < 52; k0 += 4) {
        const int kk = k0 + 2 * khalf;
        v2f a = *(const v2f*)(&sAtt[wave][nlane][kk]);
#pragma unroll
        for (int nt = 0; nt < 2; ++nt) {
            const int n = nt * 16 + nlane;
            v2f b; b.x = sV[kk][n]; b.y = sV[kk + 1][n];
            oacc[nt] = wmma4(a, b, oacc[nt]);
        }
    }

#pragma unroll
    for (int nt = 0; nt < 2; ++nt) {
#pragma unroll
        for (int r = 0; r < 8; ++r) {
            const int irow = wave * 16 + 8 * khalf + r;
            if (irow < TOK)
                O[(size_t)irow * HD + nt * 16 + nlane] = oacc[nt][r];
        }
    }
}

// =====================================================================
// Kernel 3: output projection + inverse cyclic shift.
//   grid (1568, 2) x block 128 : one 16x48 tile per wave.
// =====================================================================
__global__ __launch_bounds__(128) void outproj_kernel(
    const float* __restrict__ ows, const float* __restrict__ wout,
    const float* __restrict__ bout, float* __restrict__ out) {

    const int lane  = threadIdx.x & 31;
    const int nlane = lane & 15;
    const int khalf = lane >> 4;
    const int mt    = blockIdx.x * 4 + (threadIdx.x >> 5);
    const int nbase = blockIdx.y * 48;

    // A rows: final token t, gathered from windowed attention output
    const int t   = mt * 16 + nlane;
    const int bb  = t / (IMGH * IMGW);
    const int rem = t % (IMGH * IMGW);
    const int h = rem / IMGW, w = rem % IMGW;
    const int hsv = (h >= DISP) ? h - DISP : h + (IMGH - DISP);
    const int wsv = (w >= DISP) ? w - DISP : w + (IMGW - DISP);
    const int wr = hsv / WS7, i = hsv % WS7;
    const int wc = wsv / WS7, j = wsv % WS7;
    const int win = wr * NWH + wc;
    const int p   = i * WS7 + j;
    const float* h0 = ows + (((size_t)(bb * HEADS + 0) * NW + win) * TOK + p) * HD;
    const float* h1 = ows + (((size_t)(bb * HEADS + 1) * NW + win) * TOK + p) * HD;
    const float* h2 = ows + (((size_t)(bb * HEADS + 2) * NW + win) * TOK + p) * HD;

    v8f c0 = {}, c1 = {}, c2 = {};
#pragma unroll
    for (int k0 = 0; k0 < INNER; k0 += 4) {
        const int kk = k0 + 2 * khalf;              // pair never straddles a head
        const float* hb = (kk < 32) ? h0 : ((kk < 64) ? h1 : h2);
        v2f a = *(const v2f*)(hb + (kk & 31));
        const float* wrow = wout + (size_t)kk * HIDDEN + nbase + nlane;
        v2f b0; b0.x = wrow[0];  b0.y = wrow[HIDDEN];
        v2f b1; b1.x = wrow[16]; b1.y = wrow[HIDDEN + 16];
        v2f b2; b2.x = wrow[32]; b2.y = wrow[HIDDEN + 32];
        c0 = wmma4(a, b0, c0);
        c1 = wmma4(a, b1, c1);
        c2 = wmma4(a, b2, c2);
    }

#pragma unroll
    for (int nt = 0; nt < 3; ++nt) {
        const int col = nbase + nt * 16 + nlane;
        const float bias = bout[col];
        v8f acc = (nt == 0) ? c0 : ((nt == 1) ? c1 : c2);
#pragma unroll
        for (int r = 0; r < 8; ++r) {
            const int tok = mt * 16 + 8 * khalf + r;
            out[(size_t)tok * HIDDEN + col] = acc[r] + bias;
        }
    }
}

// =====================================================================
extern "C" void kernel_launch(void* const* d_in, const int* in_sizes, int n_in,
                              void* d_out, int out_size, void* d_ws, size_t ws_size,
                              hipStream_t stream) {
    const float* x    = (const float*)d_in[0];
    const float* wqkv = (const float*)d_in[1];
    const float* bqkv = (const float*)d_in[2];
    const float* pos  = (const float*)d_in[3];
    const float* wout = (const float*)d_in[4];
    const float* bout = (const float*)d_in[5];
    float* out = (float*)d_out;

    float* qws = (float*)d_ws;          // 38.5 MB each; O aliases qws
    float* kws = qws + PERMAT;
    float* vws = kws + PERMAT;

    dim3 g1(1568, 6), b1(128);
    qkv_kernel<<<g1, b1, 0, stream>>>(x, wqkv, bqkv, qws, kws, vws);

    dim3 g2(6144), b2(128);
    attn_kernel<<<g2, b2, 0, stream>>>(qws, kws, vws, pos, /*O=*/qws);

    dim3 g3(1568, 2), b3(128);
    outproj_kernel<<<g3, b3, 0, stream>>>(qws, wout, bout, out);
}